// SeparationLoss_39513699123324
// MI455X (gfx1250) — compile-verified
//
#include <hip/hip_runtime.h>

typedef __attribute__((ext_vector_type(2))) float v2f;
typedef __attribute__((ext_vector_type(4))) float v4f;
typedef __attribute__((ext_vector_type(8))) float v8f;

namespace {
constexpr int   kBlocks      = 512;
constexpr int   kThreads     = 256;          // 8 waves (wave32)
constexpr int   kWaves       = kThreads / 32;
constexpr int   kChunk       = 64;           // batches staged per LDS refill
constexpr int   kJoints      = 17;
constexpr int   kBatchFloats = kJoints * 3;  // 51 floats per batch
constexpr int   kChunkFloats = kChunk * kBatchFloats;   // 3264
constexpr int   kChunkVec4   = kChunkFloats / 4;        // 816
constexpr int   kFullVec4    = (kChunkVec4 / kThreads) * kThreads; // 768
constexpr int   kTailVec4    = kChunkVec4 - kFullVec4;  // 48
constexpr float kThr2        = 0.01f;        // 0.1^2
}

// Per-batch math (16x16 block via WMMA):
//   e_ij = 2*g_ij + (thr2 - n_j) - n_i ,  hinge = max(0, e_ij)
// computed as   E = (2*P) x P^T + Bias   where Bias = A2 x B2 (rank-2):
//   A2 row i = (-n_i, 1, 0, 0),  B2 col j = (1, thr2 - n_j, 0, 0)^T
// Diagonal entries evaluate exactly to thr2 -> subtract 16*thr2/batch later.
// Joint 16 (the 17th) handled scalar; lane-half duplication supplies the x2
// for ordered pairs (16,j) and (j,16).
__global__ __launch_bounds__(kThreads)
void sep_loss_main(const float* __restrict__ kps,
                   float* __restrict__ partial,
                   int nBatch)
{
  __shared__ float smem[2][kChunkFloats];   // double-buffered: 2 x 13056 B
  __shared__ float wsum[kWaves];

  const int tid  = threadIdx.x;
  const int lane = tid & 31;
  const int wave = tid >> 5;
  const int jl   = lane & 15;   // joint index 0..15 for this lane
  const int half = lane >> 4;   // 0: K=0,1 half ; 1: K=2,3 half

  const int perBlock = (nBatch + (int)gridDim.x - 1) / (int)gridDim.x;
  const int bStart   = (int)blockIdx.x * perBlock;
  const int bEnd     = min(bStart + perBlock, nBatch);

  float acc   = 0.0f;
  int   nDone = 0;

  // Issue one chunk's worth of global->LDS traffic (async path, no VGPR hop).
  auto issueChunk = [&](int c0, int bufIdx) {
    const int nb = min(kChunk, bEnd - c0);
    float* sdst = smem[bufIdx];
    const float* gsrc = kps + (size_t)c0 * kBatchFloats;
    if (nb == kChunk) {
      const uint32_t ldsBase = (uint32_t)(uintptr_t)sdst;  // low 32b = LDS offset
      #pragma unroll
      for (int i = 0; i < kFullVec4 / kThreads; ++i) {     // 3 full-EXEC issues
        const uint32_t l = ldsBase + (uint32_t)tid * 16u + (uint32_t)i * 4096u;
        const uint32_t g = (uint32_t)tid * 16u + (uint32_t)i * 4096u;
        asm volatile("global_load_async_to_lds_b128 %0, %1, %2"
                     :: "v"(l), "v"(g), "s"(gsrc) : "memory");
      }
      if (tid < kTailVec4) {                               // 48-slot tail
        ((v4f*)sdst)[kFullVec4 + tid] = ((const v4f*)gsrc)[kFullVec4 + tid];
      }
    } else {
      for (int s = tid; s < nb * kBatchFloats; s += kThreads) sdst[s] = gsrc[s];
    }
  };

  int buf = 0;
  if (bStart < bEnd) issueChunk(bStart, 0);                // prologue

  for (int c0 = bStart; c0 < bEnd; c0 += kChunk) {
    asm volatile("s_wait_asynccnt 0" ::: "memory");        // my chunk landed
    __syncthreads();                                       // everyone's landed
    if (c0 + kChunk < bEnd) issueChunk(c0 + kChunk, buf ^ 1);  // prefetch next

    const int nb  = min(kChunk, bEnd - c0);
    const int bpw = (nb + kWaves - 1) / kWaves;
    int w0 = min(wave * bpw, nb);
    int w1 = min(w0 + bpw, nb);
    w0 = __builtin_amdgcn_readfirstlane(w0);               // scalar loop bounds
    w1 = __builtin_amdgcn_readfirstlane(w1);
    nDone += (w1 - w0);

    const float* sbuf = smem[buf];
    for (int b = w0; b < w1; ++b) {       // uniform per wave: EXEC all-ones
      const float* kb = &sbuf[b * kBatchFloats];

      // Per-lane joint coords straight from LDS (x,z merge into one
      // ds_load_2addr_b32; no cross-lane traffic needed at all).
      const float x = kb[3 * jl + 0];
      const float z = kb[3 * jl + 2];
      const float y = kb[3 * jl + 1];
      const float colN = fmaf(x, x, fmaf(y, y, z * z));    // n_{jl}, both halves

      // A/B image for the Gram WMMA (identical contents for A and B):
      //   lanes 0-15 : (x_jl, y_jl)   lanes 16-31 : (z_jl, 0)
      const float ax = half ? z : x;
      const float ay = half ? 0.0f : y;

      v2f aG; aG.x = ax + ax;  aG.y = ay + ay;      // 2*P
      v2f bG; bG.x = ax;       bG.y = ay;           // P^T image

      v2f a2; a2.x = half ? 0.0f : -colN; a2.y = half ? 0.0f : 1.0f;
      v2f b2; b2.x = half ? 0.0f : 1.0f;  b2.y = half ? 0.0f : (kThr2 - colN);

      v8f cz = {0.f, 0.f, 0.f, 0.f, 0.f, 0.f, 0.f, 0.f};
      // Bias = A2 x B2 :  bias_ij = thr2 - n_i - n_j
      v8f bias = __builtin_amdgcn_wmma_f32_16x16x4_f32(
          false, a2, false, b2, (short)0, cz, false, false);
      // E = (2P) x P^T + Bias
      v8f e = __builtin_amdgcn_wmma_f32_16x16x4_f32(
          false, aG, false, bG, (short)0, bias, false, false);

      // 17th joint vs joints 0..15 (lane-half duplication -> ordered x2).
      const float x16 = kb[48], y16 = kb[49], z16 = kb[50];
      const float n16 = fmaf(x16, x16, fmaf(y16, y16, z16 * z16));
      const float dot = fmaf(x16, x, fmaf(y16, y, z16 * z));
      const float t16 = (kThr2 - n16) - colN;
      const float h16 = fmaxf(fmaf(2.0f, dot, t16), 0.0f);

      // hinge + balanced-tree sum (short dependency chains, dual-issuable)
      const float h0 = fmaxf(e[0], 0.0f), h1 = fmaxf(e[1], 0.0f);
      const float h2 = fmaxf(e[2], 0.0f), h3 = fmaxf(e[3], 0.0f);
      const float h4 = fmaxf(e[4], 0.0f), h5 = fmaxf(e[5], 0.0f);
      const float h6 = fmaxf(e[6], 0.0f), h7 = fmaxf(e[7], 0.0f);
      const float s01 = h0 + h1, s23 = h2 + h3;
      const float s45 = h4 + h5, s67 = h6 + h7;
      const float sA = s01 + s23, sB = s45 + s67;
      acc += (sA + sB) + h16;
    }
    buf ^= 1;
  }

  // Deterministic in-wave tree reduce, then per-block reduce.
  #pragma unroll
  for (int off = 16; off > 0; off >>= 1) acc += __shfl_xor(acc, off, 32);
  if (lane == 0)
    wsum[wave] = acc - (float)nDone * (16.0f * kThr2);  // remove diagonal
  __syncthreads();
  if (tid == 0) {
    float s = 0.0f;
    #pragma unroll
    for (int w = 0; w < kWaves; ++w) s += wsum[w];
    partial[blockIdx.x] = s;
  }
}

__global__ __launch_bounds__(256)
void sep_loss_reduce(const float* __restrict__ partial,
                     float* __restrict__ out,
                     int nPartial, float invB)
{
  __shared__ float s[256];
  const int tid = threadIdx.x;
  float v = 0.0f;
  for (int i = tid; i < nPartial; i += 256) v += partial[i];
  s[tid] = v;
  __syncthreads();
  #pragma unroll
  for (int off = 128; off > 0; off >>= 1) {
    if (tid < off) s[tid] += s[tid + off];
    __syncthreads();
  }
  if (tid == 0) out[0] = s[0] * invB;
}

extern "C" void kernel_launch(void* const* d_in, const int* in_sizes, int n_in,
                              void* d_out, int out_size, void* d_ws, size_t ws_size,
                              hipStream_t stream) {
  (void)n_in; (void)out_size; (void)ws_size;
  const float* kps = (const float*)d_in[0];
  const int nBatch = in_sizes[0] / kBatchFloats;   // 131072 for the reference
  float* partial = (float*)d_ws;                   // kBlocks floats of scratch

  sep_loss_main<<<kBlocks, kThreads, 0, stream>>>(kps, partial, nBatch);
  sep_loss_reduce<<<1, 256, 0, stream>>>(partial, (float*)d_out, kBlocks,
                                         1.0f / (float)nBatch);
}